// DC_CRD_85779086836063
// MI455X (gfx1250) — compile-verified
//
#include <hip/hip_runtime.h>
#include <hip/hip_bf16.h>

typedef __attribute__((ext_vector_type(2))) float v2f;
typedef __attribute__((ext_vector_type(8))) float v8f;

#define Bn 32
#define Cc 64
#define Nn 16384   // H*W = 128*128
#define NT 64      // n-columns per block in fused kernel
#define TSTR 68    // padded K-stride for transposed LDS tiles (68 % 64 banks = 4)

// ---------------- Kernel 1: global average pool -> wvec[B*C] ----------------
__global__ __launch_bounds__(256) void gap_kernel(const float* __restrict__ x,
                                                  float* __restrict__ wvec) {
    int bc = blockIdx.x;                       // b*C + c
    const float4* p = (const float4*)(x + (size_t)bc * Nn);
    float s = 0.0f;
    for (int q = threadIdx.x; q < Nn / 4; q += 256) {
        float4 v = p[q];
        s += v.x + v.y + v.z + v.w;
    }
    __shared__ float red[256];
    red[threadIdx.x] = s;
    __syncthreads();
    for (int off = 128; off > 0; off >>= 1) {
        if (threadIdx.x < off) red[threadIdx.x] += red[threadIdx.x + off];
        __syncthreads();
    }
    if (threadIdx.x == 0) wvec[bc] = red[0] * (1.0f / (float)Nn);
}

// ---------------- Kernel 2: soft adjacency A[b,i,j] ----------------
__global__ __launch_bounds__(256) void adj_kernel(const float* __restrict__ wvec,
                                                  const float* __restrict__ theta,
                                                  float* __restrict__ Aout) {
    int idx = blockIdx.x * 256 + threadIdx.x;  // b*4096 + i*64 + j
    int b  = idx >> 12;
    int ij = idx & 4095;
    int i  = ij >> 6;
    int j  = ij & 63;
    float wi = wvec[b * Cc + i];
    float wj = wvec[b * Cc + j];
    float d  = wi - wj;
    float e1 = expf(-d);
    float T1 = fabsf(1.0f - e1 / (1.0f + e1)) - 1.0f;   // T[i][j]
    float e2 = expf(d);
    float T2 = fabsf(1.0f - e2 / (1.0f + e2)) - 1.0f;   // T[j][i]
    Aout[idx] = 0.5f * (T1 + T2) * theta[ij];
}

// ---------------- Kernel 3: fused  out = W @ relu(A_b @ X) + bias ----------------
// Block: 256 threads (8 waves). blockIdx.y = batch, blockIdx.x = n-slice of NT cols.
__global__ __launch_bounds__(256) void fused_gemm_kernel(const float* __restrict__ x,
                                                         const float* __restrict__ Amat,
                                                         const float* __restrict__ W_lin,
                                                         const float* __restrict__ b_lin,
                                                         float* __restrict__ out) {
    __shared__ float A_s[Cc * Cc];      // A[b] row-major [64][64]
    __shared__ float W_s[Cc * Cc];      // W_lin row-major [64][64] (M=c, K=i)
    __shared__ float XH_t[NT * TSTR];   // transposed: XH_t[col*TSTR + k];
                                        //   stage 1: X slice; stage 2: relu(H)
    __shared__ float b_s[Cc];

    const int b  = blockIdx.y;
    const int n0 = blockIdx.x * NT;
    const int t  = threadIdx.x;

    // Load A_s and W_s: 4096 floats = 1024 float4 each; 4 per thread.
    {
        const float4* Ag = (const float4*)(Amat + (size_t)b * Cc * Cc);
        const float4* Wg = (const float4*)W_lin;
        float4* A4 = (float4*)A_s;
        float4* W4 = (float4*)W_s;
        for (int q = t; q < 1024; q += 256) {
            A4[q] = Ag[q];
            W4[q] = Wg[q];
        }
        if (t < Cc) b_s[t] = b_lin[t];
    }
    // Load X slice transposed: XH_t[col*TSTR + j] = x[b, j, n0+col].
    {
        const float* xb = x + (size_t)b * Cc * Nn + n0;
        for (int q = t; q < 1024; q += 256) {
            int row = q >> 4;          // channel j: 0..63
            int c4  = q & 15;          // float4 index along n
            float4 v = *(const float4*)(xb + (size_t)row * Nn + c4 * 4);
            int col = c4 * 4;
            XH_t[(col + 0) * TSTR + row] = v.x;
            XH_t[(col + 1) * TSTR + row] = v.y;
            XH_t[(col + 2) * TSTR + row] = v.z;
            XH_t[(col + 3) * TSTR + row] = v.w;
        }
    }
    __syncthreads();

    const int wave  = t >> 5;
    const int lane  = t & 31;
    const int lhalf = lane >> 4;   // 0: lanes 0-15, 1: lanes 16-31
    const int lmod  = lane & 15;

    // ---- Stage 1: Hmat tile = A_b @ X ; 16 tiles (mi 0..3, ns 0..3), 2 per wave ----
    v8f acc[2];
    for (int tt = 0; tt < 2; ++tt) {
        const int tile = wave * 2 + tt;
        const int mi = tile >> 2;
        const int ns = tile & 3;
        const int m  = mi * 16 + lmod;
        const int bcol = ns * 16 + lmod;
        v8f c = {};
#pragma unroll
        for (int kk = 0; kk < 16; ++kk) {
            const int k = kk * 4 + lhalf * 2;
            v2f a  = *(const v2f*)&A_s[m * Cc + k];        // contiguous K pair
            v2f bb = *(const v2f*)&XH_t[bcol * TSTR + k];  // contiguous K pair
            c = __builtin_amdgcn_wmma_f32_16x16x4_f32(false, a, false, bb,
                                                      (short)0, c, false, false);
        }
        acc[tt] = c;
    }
    __syncthreads();   // all reads of XH_t (X) complete

    // relu + write stage-1 result back into XH_t (same transposed layout, K = i)
    for (int tt = 0; tt < 2; ++tt) {
        const int tile = wave * 2 + tt;
        const int mi = tile >> 2;
        const int ns = tile & 3;
        float* hp = &XH_t[(ns * 16 + lmod) * TSTR + mi * 16 + lhalf * 8];
        float4 lo, hi;
        lo.x = acc[tt][0] > 0.0f ? acc[tt][0] : 0.0f;
        lo.y = acc[tt][1] > 0.0f ? acc[tt][1] : 0.0f;
        lo.z = acc[tt][2] > 0.0f ? acc[tt][2] : 0.0f;
        lo.w = acc[tt][3] > 0.0f ? acc[tt][3] : 0.0f;
        hi.x = acc[tt][4] > 0.0f ? acc[tt][4] : 0.0f;
        hi.y = acc[tt][5] > 0.0f ? acc[tt][5] : 0.0f;
        hi.z = acc[tt][6] > 0.0f ? acc[tt][6] : 0.0f;
        hi.w = acc[tt][7] > 0.0f ? acc[tt][7] : 0.0f;
        *(float4*)(hp)     = lo;
        *(float4*)(hp + 4) = hi;
    }
    __syncthreads();

    // ---- Stage 2: G = W_lin @ relu(H) + bias, stored transposed: out[b][n*64 + c] ----
    for (int tt = 0; tt < 2; ++tt) {
        const int tile = wave * 2 + tt;
        const int ci = tile >> 2;
        const int ns = tile & 3;
        const int m  = ci * 16 + lmod;
        const int bcol = ns * 16 + lmod;
        v8f c = {};
#pragma unroll
        for (int kk = 0; kk < 16; ++kk) {
            const int k = kk * 4 + lhalf * 2;
            v2f a  = *(const v2f*)&W_s[m * Cc + k];
            v2f bb = *(const v2f*)&XH_t[bcol * TSTR + k];
            c = __builtin_amdgcn_wmma_f32_16x16x4_f32(false, a, false, bb,
                                                      (short)0, c, false, false);
        }
        // Lane holds N = lmod, M rows {lhalf*8 + 0..7} -> 8 contiguous c-channels.
        const int n    = n0 + ns * 16 + lmod;
        const int cch0 = ci * 16 + lhalf * 8;
        float* op = out + (size_t)b * Nn * Cc + (size_t)n * Cc + cch0;
        float4 lo, hi;
        lo.x = c[0] + b_s[cch0 + 0];
        lo.y = c[1] + b_s[cch0 + 1];
        lo.z = c[2] + b_s[cch0 + 2];
        lo.w = c[3] + b_s[cch0 + 3];
        hi.x = c[4] + b_s[cch0 + 4];
        hi.y = c[5] + b_s[cch0 + 5];
        hi.z = c[6] + b_s[cch0 + 6];
        hi.w = c[7] + b_s[cch0 + 7];
        *(float4*)(op)     = lo;
        *(float4*)(op + 4) = hi;
    }
}

extern "C" void kernel_launch(void* const* d_in, const int* in_sizes, int n_in,
                              void* d_out, int out_size, void* d_ws, size_t ws_size,
                              hipStream_t stream) {
    const float* x     = (const float*)d_in[0];   // [B, C, H, W]
    const float* theta = (const float*)d_in[1];   // [C, C]
    const float* W_lin = (const float*)d_in[2];   // [C, C]
    const float* b_lin = (const float*)d_in[3];   // [C]
    float* out = (float*)d_out;

    float* wvec = (float*)d_ws;                   // B*C floats = 8 KB
    float* Amat = wvec + Bn * Cc;                 // B*C*C floats = 512 KB

    gap_kernel<<<Bn * Cc, 256, 0, stream>>>(x, wvec);
    adj_kernel<<<(Bn * Cc * Cc) / 256, 256, 0, stream>>>(wvec, theta, Amat);

    dim3 grid(Nn / NT, Bn);
    fused_gemm_kernel<<<grid, 256, 0, stream>>>(x, Amat, W_lin, b_lin, out);
}